// HGT_41059887350266
// MI455X (gfx1250) — compile-verified
//
#include <hip/hip_runtime.h>
#include <math.h>

// ---- problem constants (from reference) ----
#define NTY   2
#define NED   2
#define NN    50000
#define FINW  128
#define HIDW  128
#define NH    4
#define DDIM  32
#define NLAY  2
#define NEDGE 400000
// EDGE_SRC[e] = e, EDGE_DST[e] = 1-e

typedef __attribute__((ext_vector_type(2))) float v2f;
typedef __attribute__((ext_vector_type(8))) float v8f;
typedef int v4i_ __attribute__((vector_size(16)));              // matches builtin param
typedef __attribute__((address_space(3))) v4i_* lds_v4i_p;      // LDS-qualified dest

// gfx1250 async global->LDS copy path (ASYNCcnt), guarded so either toolchain compiles
#if defined(__has_builtin)
#if __has_builtin(__builtin_amdgcn_global_load_async_to_lds_b128) && \
    __has_builtin(__builtin_amdgcn_s_wait_asynccnt)
#define USE_ASYNC_LDS 1
#endif
#endif
#ifndef USE_ASYNC_LDS
#define USE_ASYNC_LDS 0
#endif

__device__ __forceinline__ float gelu_exact(float x) {
    return 0.5f * x * (1.0f + erff(x * 0.70710678118654752f));
}
__device__ __forceinline__ float sigmoidf_(float x) {
    return 1.0f / (1.0f + expf(-x));
}
// deterministic float atomic max via sign-magnitude int trick
__device__ __forceinline__ void atomicMaxFloat(float* addr, float val) {
    if (val >= 0.0f) atomicMax((int*)addr, __float_as_int(val));
    else             atomicMin((unsigned int*)addr, __float_as_uint(val));
}

// ============================================================================
// GEMM: C[n_rows,128] = act_out( A[n_rows,128] @ W[128,128] + bias )
// mode 0: out = relu(...)        (input projection)
// mode 1: out = ...              (q / k_rel / v_rel projections)
// mode 2: h = beta*out + (1-beta)*h, beta = sigmoid(skip[idx])  (A pre-GELUed)
// V_WMMA_F32_16X16X4_F32; 256 threads = 8 waves; block tile 64x128; each wave
// computes 16x64 (4 accumulators) -> 4 WMMA per 1 LDS A-fragment per k-step.
// A tile staged via GLOBAL_LOAD_ASYNC_TO_LDS_B128 when available.
// ============================================================================
#define TILE_M 64
#define ASTRIDE 132   // 128 + 4 pad -> conflict-free A-fragment reads

__global__ __launch_bounds__(256)
void gemm128_kernel(const float* __restrict__ A, const float* __restrict__ W,
                    const float* __restrict__ bias, float* __restrict__ C,
                    int n_rows, int mode,
                    const float* __restrict__ skip, int skip_idx)
{
    __shared__ float As[TILE_M * ASTRIDE];
    const int tid = threadIdx.x;
    const int rowBlock = blockIdx.x * TILE_M;

    // ---- stage A tile (64x128), 2048 float4 slots, 8 per thread ----
    #pragma unroll
    for (int i = 0; i < 8; ++i) {
        int idx  = tid + i * 256;
        int r    = idx >> 5;               // 32 float4 per row
        int c4   = idx & 31;
        int grow = rowBlock + r;
        float* dstp = As + r * ASTRIDE + c4 * 4;
#if USE_ASYNC_LDS
        if (grow < n_rows) {
            __builtin_amdgcn_global_load_async_to_lds_b128(
                (v4i_*)const_cast<float*>(A + (size_t)grow * HIDW + c4 * 4),
                (lds_v4i_p)dstp, 0, 0);
        } else {
            dstp[0] = 0.f; dstp[1] = 0.f; dstp[2] = 0.f; dstp[3] = 0.f;
        }
#else
        float4 v = make_float4(0.f, 0.f, 0.f, 0.f);
        if (grow < n_rows)
            v = ((const float4*)(A + (size_t)grow * HIDW))[c4];
        dstp[0] = v.x; dstp[1] = v.y; dstp[2] = v.z; dstp[3] = v.w;
#endif
    }
#if USE_ASYNC_LDS
    __builtin_amdgcn_s_wait_asynccnt(0);
#endif
    __syncthreads();

    const int lane    = tid & 31;
    const int wave    = tid >> 5;
    const int rowTile = (wave >> 1) << 4;   // 0,16,32,48
    const int colBase = (wave & 1) << 6;    // 0,64
    const int mrow    = lane & 15;
    const int khalf   = (lane >> 4) << 1;   // lanes16-31 hold K+2,K+3

    v8f c0 = {}, c1 = {}, c2 = {}, c3 = {};
    const float* arow = As + (rowTile + mrow) * ASTRIDE;

    #pragma unroll 2
    for (int k0 = 0; k0 < 128; k0 += 4) {
        const int kb = k0 + khalf;
        v2f a; a.x = arow[kb]; a.y = arow[kb + 1];
        // B 4x16 fragments: lane n -> column colBase+n+{0,16,32,48}
        const float* w0 = W + (size_t)kb * HIDW + colBase + mrow;
        v2f b0, b1, b2, b3;
        b0.x = w0[0];   b0.y = w0[HIDW];
        b1.x = w0[16];  b1.y = w0[HIDW + 16];
        b2.x = w0[32];  b2.y = w0[HIDW + 32];
        b3.x = w0[48];  b3.y = w0[HIDW + 48];
        c0 = __builtin_amdgcn_wmma_f32_16x16x4_f32(false, a, false, b0, (short)0, c0, false, false);
        c1 = __builtin_amdgcn_wmma_f32_16x16x4_f32(false, a, false, b1, (short)0, c1, false, false);
        c2 = __builtin_amdgcn_wmma_f32_16x16x4_f32(false, a, false, b2, (short)0, c2, false, false);
        c3 = __builtin_amdgcn_wmma_f32_16x16x4_f32(false, a, false, b3, (short)0, c3, false, false);
    }

    float beta = 1.f, omb = 0.f;
    if (mode == 2) { beta = sigmoidf_(skip[skip_idx]); omb = 1.f - beta; }

    const int rowOff = (lane >> 4) << 3;    // C layout: VGPR r -> M = r (+8 upper half)
    const int col0 = colBase + mrow;
    const float bi0 = bias[col0], bi1 = bias[col0 + 16];
    const float bi2 = bias[col0 + 32], bi3 = bias[col0 + 48];
    #pragma unroll
    for (int r = 0; r < 8; ++r) {
        int grow = rowBlock + rowTile + rowOff + r;
        if (grow >= n_rows) continue;
        float v0 = c0[r] + bi0, v1 = c1[r] + bi1;
        float v2 = c2[r] + bi2, v3 = c3[r] + bi3;
        if (mode == 0) {
            v0 = fmaxf(v0, 0.f); v1 = fmaxf(v1, 0.f);
            v2 = fmaxf(v2, 0.f); v3 = fmaxf(v3, 0.f);
        }
        float* crow = C + (size_t)grow * HIDW;
        if (mode == 2) {
            crow[col0]      = beta * v0 + omb * crow[col0];
            crow[col0 + 16] = beta * v1 + omb * crow[col0 + 16];
            crow[col0 + 32] = beta * v2 + omb * crow[col0 + 32];
            crow[col0 + 48] = beta * v3 + omb * crow[col0 + 48];
        } else {
            crow[col0]      = v0;
            crow[col0 + 16] = v1;
            crow[col0 + 32] = v2;
            crow[col0 + 48] = v3;
        }
    }
}

// ============================================================================
// Fold rel_att/rel_msg (block-diag per head) into projection weights + bias.
// w_eff[t][i][h*32+dp] = sum_d w[t][i][h*32+d] * rel[t(=edge)][h][d][dp]
// i==128 slot computes effective bias.
// ============================================================================
__global__ void effw_kernel(const float* __restrict__ w,   // + l*NT*HID*HID
                            const float* __restrict__ b,   // + l*NT*HID
                            const float* __restrict__ rel, // + l*NE*H*D*D
                            float* __restrict__ w_eff, float* __restrict__ b_eff)
{
    int g = blockIdx.x * blockDim.x + threadIdx.x;
    if (g >= NTY * 129 * NH * DDIM) return;
    int dp = g % DDIM;
    int hh = (g / DDIM) % NH;
    int i  = (g / (DDIM * NH)) % 129;
    int t  = g / (DDIM * NH * 129);
    const float* R = rel + ((size_t)t * NH + hh) * DDIM * DDIM;  // edge e = t
    float acc = 0.f;
    if (i < HIDW) {
        const float* wr = w + ((size_t)t * HIDW + i) * HIDW + hh * DDIM;
        for (int d = 0; d < DDIM; ++d) acc += wr[d] * R[d * DDIM + dp];
        w_eff[((size_t)t * HIDW + i) * HIDW + hh * DDIM + dp] = acc;
    } else {
        const float* br = b + (size_t)t * HIDW + hh * DDIM;
        for (int d = 0; d < DDIM; ++d) acc += br[d] * R[d * DDIM + dp];
        b_eff[(size_t)t * HIDW + hh * DDIM + dp] = acc;
    }
}

// ---- "last write wins" edge-weight scaling -------------------------------
__global__ void lastedge_kernel(const int* __restrict__ ei, int* __restrict__ last) {
    int g = blockIdx.x * blockDim.x + threadIdx.x;
    if (g >= NED * NEDGE) return;
    int e = g / NEDGE, j = g % NEDGE;
    int sn = ei[(size_t)e * 2 * NEDGE + j];           // src row of edge type e
    atomicMax(&last[e * NN + sn], j);                 // max index == last in order
}
__global__ void applyw_kernel(const float* __restrict__ ew, const int* __restrict__ last,
                              float* __restrict__ h) {
    int g = blockIdx.x * blockDim.x + threadIdx.x;
    if (g >= NED * NN * 32) return;
    int c4 = g & 31;
    int i  = (g >> 5) % NN;
    int e  = g / (NN * 32);
    int le = last[e * NN + i];
    if (le < 0) return;
    float w = sigmoidf_(ew[(size_t)e * NEDGE + le]);
    float4* p = (float4*)(h + ((size_t)e * NN + i) * HIDW) + c4;  // src type == e
    float4 v = *p;
    v.x *= w; v.y *= w; v.z *= w; v.w *= w;
    *p = v;
}

// ---- edge attention: 8 lanes per (edge,head), float4 gathers -------------
__global__ __launch_bounds__(256)
void scores_kernel(const float* __restrict__ q_t, const float* __restrict__ k_s,
                   const int* __restrict__ src, const int* __restrict__ dst,
                   const float* __restrict__ relp, float* __restrict__ scores,
                   float* __restrict__ mmax)
{
    int g = blockIdx.x * blockDim.x + threadIdx.x;
    int item = g >> 3, sub = g & 7;
    if (item >= NEDGE * NH) return;
    int edge = item >> 2, hh = item & 3;
    int sn = src[edge], dn = dst[edge];
    float4 qv = *(const float4*)(q_t + (size_t)dn * HIDW + hh * DDIM + sub * 4);
    float4 kv = *(const float4*)(k_s + (size_t)sn * HIDW + hh * DDIM + sub * 4);
    float d = qv.x * kv.x + qv.y * kv.y + qv.z * kv.z + qv.w * kv.w;
    d += __shfl_xor(d, 1);
    d += __shfl_xor(d, 2);
    d += __shfl_xor(d, 4);
    if (sub == 0) {
        float s = d * relp[hh] * 0.17677669529663687f;  // 1/sqrt(32)
        scores[item] = s;
        atomicMaxFloat(&mmax[(size_t)dn * NH + hh], s);
    }
}

__global__ void expsum_kernel(float* __restrict__ scores, const int* __restrict__ dst,
                              const float* __restrict__ mmax, float* __restrict__ ssum)
{
    int item = blockIdx.x * blockDim.x + threadIdx.x;
    if (item >= NEDGE * NH) return;
    int edge = item >> 2, hh = item & 3;
    int dn = dst[edge];
    float e = expf(scores[item] - mmax[(size_t)dn * NH + hh]);
    scores[item] = e;
    atomicAdd(&ssum[(size_t)dn * NH + hh], e);
}

__global__ __launch_bounds__(256)
void scatter_kernel(const float* __restrict__ scores, const float* __restrict__ ssum,
                    const float* __restrict__ v_s, const int* __restrict__ src,
                    const int* __restrict__ dst, float* __restrict__ agg_t)
{
    int g = blockIdx.x * blockDim.x + threadIdx.x;
    int item = g >> 3, sub = g & 7;
    if (item >= NEDGE * NH) return;
    int edge = item >> 2, hh = item & 3;
    int sn = src[edge], dn = dst[edge];
    float alpha = scores[item] / (ssum[(size_t)dn * NH + hh] + 1e-16f);
    float4 vv = *(const float4*)(v_s + (size_t)sn * HIDW + hh * DDIM + sub * 4);
    float* ap = agg_t + (size_t)dn * HIDW + hh * DDIM + sub * 4;
    atomicAdd(ap + 0, vv.x * alpha);
    atomicAdd(ap + 1, vv.y * alpha);
    atomicAdd(ap + 2, vv.z * alpha);
    atomicAdd(ap + 3, vv.w * alpha);
}

// ---- elementwise GELU pre-pass (keeps GEMM staging a pure async copy) ----
__global__ void gelu_kernel(float* __restrict__ p, size_t n) {
    size_t i = (size_t)blockIdx.x * blockDim.x + threadIdx.x;
    if (i < n) p[i] = gelu_exact(p[i]);
}

// ---- fills ---------------------------------------------------------------
__global__ void fillf_kernel(float* p, float v, size_t n) {
    size_t i = (size_t)blockIdx.x * blockDim.x + threadIdx.x;
    if (i < n) p[i] = v;
}
__global__ void filli_kernel(int* p, int v, size_t n) {
    size_t i = (size_t)blockIdx.x * blockDim.x + threadIdx.x;
    if (i < n) p[i] = v;
}

static inline int nblk(size_t n, int bs) { return (int)((n + (size_t)bs - 1) / (size_t)bs); }

extern "C" void kernel_launch(void* const* d_in, const int* in_sizes, int n_in,
                              void* d_out, int out_size, void* d_ws, size_t ws_size,
                              hipStream_t stream)
{
    const float* x       = (const float*)d_in[0];
    const int*   ei      = (const int*)  d_in[1];
    const float* ew      = (const float*)d_in[2];
    const float* lin_w   = (const float*)d_in[3];
    const float* lin_b   = (const float*)d_in[4];
    const float* k_w     = (const float*)d_in[5];
    const float* k_b     = (const float*)d_in[6];
    const float* q_w     = (const float*)d_in[7];
    const float* q_b     = (const float*)d_in[8];
    const float* v_w     = (const float*)d_in[9];
    const float* v_b     = (const float*)d_in[10];
    const float* a_w     = (const float*)d_in[11];
    const float* a_b     = (const float*)d_in[12];
    const float* skip    = (const float*)d_in[13];
    const float* rel_att = (const float*)d_in[14];
    const float* rel_msg = (const float*)d_in[15];
    const float* rel_p   = (const float*)d_in[16];

    float* h = (float*)d_out;  // [NT, N, HID] lives in d_out throughout

    // workspace layout (~221 MB)
    const size_t per = (size_t)NTY * NN * HIDW;      // 12.8M floats
    float* ws    = (float*)d_ws;
    float* q     = ws;
    float* krel  = q    + per;
    float* vrel  = krel + per;
    float* agg   = vrel + per;
    float* sco   = agg  + per;                        // NE*E*H
    float* mmax  = sco  + (size_t)NED * NEDGE * NH;   // NT*N*H
    float* ssum  = mmax + (size_t)NTY * NN * NH;
    float* kweff = ssum + (size_t)NTY * NN * NH;      // NT*128*128
    float* vweff = kweff + (size_t)NTY * HIDW * HIDW;
    float* kbeff = vweff + (size_t)NTY * HIDW * HIDW; // NT*128
    float* vbeff = kbeff + (size_t)NTY * HIDW;
    int*   lastE = (int*)(vbeff + (size_t)NTY * HIDW);// NE*N ints

    const int gB = nblk(NN, TILE_M);

    // 1) input projection + ReLU (WMMA fp32)
    for (int t = 0; t < NTY; ++t)
        gemm128_kernel<<<gB, 256, 0, stream>>>(
            x + (size_t)t * NN * FINW, lin_w + (size_t)t * FINW * HIDW,
            lin_b + (size_t)t * HIDW, h + (size_t)t * NN * HIDW, NN, 0, nullptr, 0);

    // 2) sigmoid edge-weight scaling, last-write-wins per source node
    filli_kernel<<<nblk((size_t)NED * NN, 256), 256, 0, stream>>>(lastE, -1, (size_t)NED * NN);
    lastedge_kernel<<<nblk((size_t)NED * NEDGE, 256), 256, 0, stream>>>(ei, lastE);
    applyw_kernel<<<nblk((size_t)NED * NN * 32, 256), 256, 0, stream>>>(ew, lastE, h);

    for (int l = 0; l < NLAY; ++l) {
        // 3) fold rel matrices into k/v projection weights
        const size_t nEff = (size_t)NTY * 129 * NH * DDIM;
        effw_kernel<<<nblk(nEff, 256), 256, 0, stream>>>(
            k_w + (size_t)l * NTY * HIDW * HIDW, k_b + (size_t)l * NTY * HIDW,
            rel_att + (size_t)l * NED * NH * DDIM * DDIM, kweff, kbeff);
        effw_kernel<<<nblk(nEff, 256), 256, 0, stream>>>(
            v_w + (size_t)l * NTY * HIDW * HIDW, v_b + (size_t)l * NTY * HIDW,
            rel_msg + (size_t)l * NED * NH * DDIM * DDIM, vweff, vbeff);

        // 4) q / k_rel / v_rel projections (WMMA fp32)
        for (int t = 0; t < NTY; ++t) {
            const float* hp = h + (size_t)t * NN * HIDW;
            gemm128_kernel<<<gB, 256, 0, stream>>>(
                hp, q_w + ((size_t)l * NTY + t) * HIDW * HIDW,
                q_b + ((size_t)l * NTY + t) * HIDW,
                q + (size_t)t * NN * HIDW, NN, 1, nullptr, 0);
            gemm128_kernel<<<gB, 256, 0, stream>>>(
                hp, kweff + (size_t)t * HIDW * HIDW, kbeff + (size_t)t * HIDW,
                krel + (size_t)t * NN * HIDW, NN, 1, nullptr, 0);
            gemm128_kernel<<<gB, 256, 0, stream>>>(
                hp, vweff + (size_t)t * HIDW * HIDW, vbeff + (size_t)t * HIDW,
                vrel + (size_t)t * NN * HIDW, NN, 1, nullptr, 0);
        }

        // 5) clear accumulators / softmax state
        fillf_kernel<<<nblk(per, 256), 256, 0, stream>>>(agg, 0.f, per);
        fillf_kernel<<<nblk((size_t)NTY * NN * NH, 256), 256, 0, stream>>>(ssum, 0.f, (size_t)NTY * NN * NH);
        fillf_kernel<<<nblk((size_t)NTY * NN * NH, 256), 256, 0, stream>>>(mmax, -3.402823466e38f, (size_t)NTY * NN * NH);

        // 6) edge attention: scores -> segment softmax -> scatter messages
        for (int e = 0; e < NED; ++e) {
            const int t = 1 - e, s = e;  // EDGE_SRC[e]=e, EDGE_DST[e]=1-e
            const int* srcp = ei + (size_t)e * 2 * NEDGE;
            const int* dstp = srcp + NEDGE;
            float* sc = sco + (size_t)e * NEDGE * NH;
            const size_t n8 = (size_t)NEDGE * NH * 8;
            scores_kernel<<<nblk(n8, 256), 256, 0, stream>>>(
                q + (size_t)t * NN * HIDW, krel + (size_t)s * NN * HIDW,
                srcp, dstp, rel_p + ((size_t)l * NED + e) * NH,
                sc, mmax + (size_t)t * NN * NH);
            expsum_kernel<<<nblk((size_t)NEDGE * NH, 256), 256, 0, stream>>>(
                sc, dstp, mmax + (size_t)t * NN * NH, ssum + (size_t)t * NN * NH);
            scatter_kernel<<<nblk(n8, 256), 256, 0, stream>>>(
                sc, ssum + (size_t)t * NN * NH, vrel + (size_t)s * NN * HIDW,
                srcp, dstp, agg + (size_t)t * NN * HIDW);
        }

        // 7) output projection: h = beta*(gelu(agg)@a_w + a_b) + (1-beta)*h
        gelu_kernel<<<nblk(per, 256), 256, 0, stream>>>(agg, per);
        for (int t = 0; t < NTY; ++t)
            gemm128_kernel<<<gB, 256, 0, stream>>>(
                agg + (size_t)t * NN * HIDW, a_w + ((size_t)l * NTY + t) * HIDW * HIDW,
                a_b + ((size_t)l * NTY + t) * HIDW,
                h + (size_t)t * NN * HIDW, NN, 2, skip, l * NTY + t);
    }
}